// GraphSAGEAttackPredictor_64570538328559
// MI455X (gfx1250) — compile-verified
//
#include <hip/hip_runtime.h>
#include <hip/hip_bf16.h>

// ---------------------------------------------------------------------------
// GraphSAGE (3x SAGEConv + BN/ReLU) for MI455X / gfx1250, fp32 WMMA path.
// Round 3: 2x4 register-tiled WMMA GEMM (32x64 per wave32) -- spill-free
// (8 accumulators = 64 VGPRs) with __launch_bounds__ to free the allocator.
// ---------------------------------------------------------------------------

typedef __attribute__((ext_vector_type(2))) float v2f;
typedef __attribute__((ext_vector_type(8))) float v8f;

#define N_NODES 50000
#define N_EDGES 1600000

// ---------------------------------------------------------------------------
// Degree: one float atomic per edge.
// ---------------------------------------------------------------------------
__global__ void deg_kernel(const int* __restrict__ dst, float* __restrict__ deg,
                           int nedges) {
    int e = blockIdx.x * blockDim.x + threadIdx.x;
    if (e < nedges) atomicAdd(&deg[dst[e]], 1.0f);
}

__global__ void invdeg_kernel(const float* __restrict__ deg,
                              float* __restrict__ invdeg, int n) {
    int i = blockIdx.x * blockDim.x + threadIdx.x;
    if (i < n) invdeg[i] = 1.0f / fmaxf(deg[i], 1.0f);
}

// ---------------------------------------------------------------------------
// Edge scatter: agg[dst] += X[src].  (K>>2) consecutive threads cooperate on
// one edge -> coalesced float4 gather of the source row, f32 atomics into the
// destination row (L2 atomic units; feature matrices are L2-resident at
// 25-51MB vs 192MB L2).  chunkShift = log2(K/4) to avoid div/mod.
// ---------------------------------------------------------------------------
__global__ void scatter_add_kernel(const float* __restrict__ X,
                                   const int* __restrict__ src,
                                   const int* __restrict__ dst,
                                   float* __restrict__ agg,
                                   int K, int chunkShift, int nedges) {
    long long idx = (long long)blockIdx.x * blockDim.x + threadIdx.x;
    long long e = idx >> chunkShift;
    int c = (int)(idx & ((1 << chunkShift) - 1));
    if (e >= nedges) return;
    int s = src[e];
    int d = dst[e];
    const float4 v = ((const float4*)(X + (size_t)s * K))[c];
    float* out = agg + (size_t)d * K + (size_t)c * 4;
    atomicAdd(out + 0, v.x);
    atomicAdd(out + 1, v.y);
    atomicAdd(out + 2, v.z);
    atomicAdd(out + 3, v.w);
}

// ---------------------------------------------------------------------------
// Fused SAGE GEMM using V_WMMA_F32_16X16X4_F32 (exact fp32):
//   H[m,n] = sum_k agg[m,k]*invdeg[m]*Wl[k,n] + sum_k X[m,k]*Wr[k,n] + bias[n]
// One wave32 computes a 32x64 output block (2x4 grid of 16x16 WMMA tiles);
// 8 f32 accumulators (64 VGPRs) => no scratch spills.  Degree normalization
// folded into the pass-1 A-operand load.
//
// A layout (16x4 f32, ISA 7.12.2): lane L, m = L&15, kbase = 2*(L>>4):
//   vgpr0 = A[m][k+kbase], vgpr1 = A[m][k+kbase+1]
// B layout (4x16): lane L, n = L&15, kbase = 2*(L>>4):
//   vgpr0 = B[k+kbase][n], vgpr1 = B[k+kbase+1][n]
// C/D layout: vgpr v: lanes 0-15 -> M=v, lanes 16-31 -> M=v+8, N = L&15.
// ---------------------------------------------------------------------------
__global__ __launch_bounds__(256, 1)
void sage_gemm_kernel(const float* __restrict__ agg,
                      const float* __restrict__ X,
                      const float* __restrict__ invdeg,
                      const float* __restrict__ Wl,
                      const float* __restrict__ Wr,
                      const float* __restrict__ bias,
                      float* __restrict__ H,
                      int K, int Nout, int tilesM) {
    const int wave   = (int)((blockIdx.x * (long long)blockDim.x + threadIdx.x) >> 5);
    const int lane   = threadIdx.x & 31;
    const int superN = Nout >> 6;                 // groups of 4 n-tiles
    const int superM = (tilesM + 1) >> 1;         // groups of 2 m-tiles
    const int sm = wave / superN;
    const int sn = wave - sm * superN;
    if (sm >= superM) return;                     // wave-uniform (EXEC all-1s)

    const int half = lane >> 4;                   // 0 or 1
    const int l15  = lane & 15;

    // 2 M-subtiles: row pointers (clamped for the ragged last supertile so
    // loads stay in-bounds; stores are skipped for invalid subtiles).
    const float* arow[2];
    const float* xrow[2];
    float idg[2];
    bool  mval[2];
#pragma unroll
    for (int i = 0; i < 2; ++i) {
        const int tm = sm * 2 + i;
        mval[i] = (tm < tilesM);
        const int m = mval[i] ? (tm * 16 + l15) : l15;
        arow[i] = agg + (size_t)m * K;
        xrow[i] = X   + (size_t)m * K;
        idg[i]  = invdeg[m];
    }
    // 4 N-subtiles: column index per lane.
    int ncol[4];
#pragma unroll
    for (int j = 0; j < 4; ++j) ncol[j] = (sn * 4 + j) * 16 + l15;

    v8f acc[2][4];
#pragma unroll
    for (int i = 0; i < 2; ++i)
#pragma unroll
        for (int j = 0; j < 4; ++j) acc[i][j] = (v8f){};

    // ---- Pass 1: normalized aggregate @ Wl ----
    for (int k = 0; k < K; k += 4) {
        const int ka = k + half * 2;
        v2f b[4];
#pragma unroll
        for (int j = 0; j < 4; ++j) {
            b[j].x = Wl[(size_t)ka * Nout + ncol[j]];
            b[j].y = Wl[(size_t)(ka + 1) * Nout + ncol[j]];
        }
#pragma unroll
        for (int i = 0; i < 2; ++i) {
            v2f a;
            a.x = arow[i][ka]     * idg[i];
            a.y = arow[i][ka + 1] * idg[i];
#pragma unroll
            for (int j = 0; j < 4; ++j)
                acc[i][j] = __builtin_amdgcn_wmma_f32_16x16x4_f32(
                    false, a, false, b[j], (short)0, acc[i][j], false, false);
        }
    }
    // ---- Pass 2: root features @ Wr ----
    for (int k = 0; k < K; k += 4) {
        const int ka = k + half * 2;
        v2f b[4];
#pragma unroll
        for (int j = 0; j < 4; ++j) {
            b[j].x = Wr[(size_t)ka * Nout + ncol[j]];
            b[j].y = Wr[(size_t)(ka + 1) * Nout + ncol[j]];
        }
#pragma unroll
        for (int i = 0; i < 2; ++i) {
            v2f a;
            a.x = xrow[i][ka];
            a.y = xrow[i][ka + 1];
#pragma unroll
            for (int j = 0; j < 4; ++j)
                acc[i][j] = __builtin_amdgcn_wmma_f32_16x16x4_f32(
                    false, a, false, b[j], (short)0, acc[i][j], false, false);
        }
    }

    // ---- Store: bias add fused; C/D layout M = v + 8*half ----
#pragma unroll
    for (int i = 0; i < 2; ++i) {
        if (!mval[i]) continue;                   // wave-uniform
        const int mbase = (sm * 2 + i) * 16 + half * 8;
#pragma unroll
        for (int j = 0; j < 4; ++j) {
            const float bn = bias[ncol[j]];
#pragma unroll
            for (int v = 0; v < 8; ++v) {
                H[(size_t)(mbase + v) * Nout + ncol[j]] = acc[i][j][v] + bn;
            }
        }
    }
}

// ---------------------------------------------------------------------------
// BatchNorm: column sums / sums-of-squares (256 threads = 256 columns,
// coalesced row-major sweep, one atomic pair per block per column).
// ---------------------------------------------------------------------------
__global__ void bn_stats_kernel(const float* __restrict__ H,
                                float* __restrict__ sums,
                                float* __restrict__ sumsq,
                                int nrows, int C, int rowsPerBlock) {
    int c  = threadIdx.x;                  // blockDim.x == C
    int r0 = blockIdx.x * rowsPerBlock;
    int r1 = r0 + rowsPerBlock;
    if (r1 > nrows) r1 = nrows;
    float s = 0.0f, s2 = 0.0f;
    for (int r = r0; r < r1; ++r) {
        float v = H[(size_t)r * C + c];
        s  += v;
        s2 += v * v;
    }
    atomicAdd(&sums[c], s);
    atomicAdd(&sumsq[c], s2);
}

__global__ void bn_finalize_kernel(const float* __restrict__ sums,
                                   const float* __restrict__ sumsq,
                                   const float* __restrict__ gamma,
                                   const float* __restrict__ beta,
                                   float* __restrict__ scale,
                                   float* __restrict__ shift,
                                   int C, float invN) {
    int c = blockIdx.x * blockDim.x + threadIdx.x;
    if (c < C) {
        float mu  = sums[c] * invN;
        float var = sumsq[c] * invN - mu * mu;     // jnp.var: 1/N
        float sc  = gamma[c] * rsqrtf(var + 1e-5f);
        scale[c] = sc;
        shift[c] = beta[c] - mu * sc;
    }
}

__global__ void bn_apply_relu_kernel(float* __restrict__ H,
                                     const float* __restrict__ scale,
                                     const float* __restrict__ shift,
                                     int total, int Cmask) {
    int i = blockIdx.x * blockDim.x + threadIdx.x;
    if (i >= total) return;
    int c = i & Cmask;                     // C is a power of two
    float v = H[i] * scale[c] + shift[c];
    H[i] = v > 0.0f ? v : 0.0f;
}

// ---------------------------------------------------------------------------
// Host-side orchestration (graph-capture safe: async memsets + kernels only).
// ---------------------------------------------------------------------------
extern "C" void kernel_launch(void* const* d_in, const int* in_sizes, int n_in,
                              void* d_out, int out_size, void* d_ws, size_t ws_size,
                              hipStream_t stream) {
    const float* x   = (const float*)d_in[0];              // [N, 128]
    const int* ei    = (const int*)d_in[1];                // [2, E]
    const int* src   = ei;
    const int* dst   = ei + N_EDGES;
    const float* Wl0 = (const float*)d_in[2];
    const float* bl0 = (const float*)d_in[3];
    const float* Wr0 = (const float*)d_in[4];
    const float* Wl1 = (const float*)d_in[5];
    const float* bl1 = (const float*)d_in[6];
    const float* Wr1 = (const float*)d_in[7];
    const float* Wl2 = (const float*)d_in[8];
    const float* bl2 = (const float*)d_in[9];
    const float* Wr2 = (const float*)d_in[10];
    const float* g0  = (const float*)d_in[11];
    const float* be0 = (const float*)d_in[12];
    const float* g1  = (const float*)d_in[13];
    const float* be1 = (const float*)d_in[14];

    // Workspace carving (floats)
    float* ws = (float*)d_ws;
    size_t off = 0;
    float* agg    = ws + off; off += (size_t)N_NODES * 256;   // aggregate buffer
    float* hA     = ws + off; off += (size_t)N_NODES * 256;   // layer-0 output
    float* hB     = ws + off; off += (size_t)N_NODES * 256;   // layer-1 output
    float* deg    = ws + off; off += N_NODES;
    float* invdeg = ws + off; off += N_NODES;
    float* sums   = ws + off; off += 256;
    float* sumsq  = ws + off; off += 256;
    float* scale  = ws + off; off += 256;
    float* shift  = ws + off; off += 256;

    const int TPB = 256;

    // ---- degrees (shared by all layers) ----
    hipMemsetAsync(deg, 0, N_NODES * sizeof(float), stream);
    deg_kernel<<<(N_EDGES + TPB - 1) / TPB, TPB, 0, stream>>>(dst, deg, N_EDGES);
    invdeg_kernel<<<(N_NODES + TPB - 1) / TPB, TPB, 0, stream>>>(deg, invdeg, N_NODES);

    const int tilesM = N_NODES / 16;        // 3125 (exact)
    const int superM = (tilesM + 1) / 2;    // 1563

    // ================= Layer 0: 128 -> 256 =================
    {
        const int K = 128, NO = 256;
        hipMemsetAsync(agg, 0, (size_t)N_NODES * K * sizeof(float), stream);
        long long sthreads = (long long)N_EDGES * (K / 4);
        scatter_add_kernel<<<(int)((sthreads + TPB - 1) / TPB), TPB, 0, stream>>>(
            x, src, dst, agg, K, 5 /* log2(128/4) */, N_EDGES);

        int waves = superM * (NO / 64);
        sage_gemm_kernel<<<(waves + 7) / 8, TPB, 0, stream>>>(
            agg, x, invdeg, Wl0, Wr0, bl0, hA, K, NO, tilesM);

        hipMemsetAsync(sums, 0, 2 * 256 * sizeof(float), stream);
        bn_stats_kernel<<<200, 256, 0, stream>>>(hA, sums, sumsq, N_NODES, NO, 250);
        bn_finalize_kernel<<<1, 256, 0, stream>>>(sums, sumsq, g0, be0, scale, shift,
                                                  NO, 1.0f / (float)N_NODES);
        int total = N_NODES * NO;
        bn_apply_relu_kernel<<<(total + TPB - 1) / TPB, TPB, 0, stream>>>(
            hA, scale, shift, total, NO - 1);
    }

    // ================= Layer 1: 256 -> 256 =================
    {
        const int K = 256, NO = 256;
        hipMemsetAsync(agg, 0, (size_t)N_NODES * K * sizeof(float), stream);
        long long sthreads = (long long)N_EDGES * (K / 4);
        scatter_add_kernel<<<(int)((sthreads + TPB - 1) / TPB), TPB, 0, stream>>>(
            hA, src, dst, agg, K, 6 /* log2(256/4) */, N_EDGES);

        int waves = superM * (NO / 64);
        sage_gemm_kernel<<<(waves + 7) / 8, TPB, 0, stream>>>(
            agg, hA, invdeg, Wl1, Wr1, bl1, hB, K, NO, tilesM);

        hipMemsetAsync(sums, 0, 2 * 256 * sizeof(float), stream);
        bn_stats_kernel<<<200, 256, 0, stream>>>(hB, sums, sumsq, N_NODES, NO, 250);
        bn_finalize_kernel<<<1, 256, 0, stream>>>(sums, sumsq, g1, be1, scale, shift,
                                                  NO, 1.0f / (float)N_NODES);
        int total = N_NODES * NO;
        bn_apply_relu_kernel<<<(total + TPB - 1) / TPB, TPB, 0, stream>>>(
            hB, scale, shift, total, NO - 1);
    }

    // ================= Layer 2: 256 -> 64 =================
    {
        const int K = 256, NO = 64;
        hipMemsetAsync(agg, 0, (size_t)N_NODES * K * sizeof(float), stream);
        long long sthreads = (long long)N_EDGES * (K / 4);
        scatter_add_kernel<<<(int)((sthreads + TPB - 1) / TPB), TPB, 0, stream>>>(
            hB, src, dst, agg, K, 6 /* log2(256/4) */, N_EDGES);

        int waves = superM * (NO / 64);
        sage_gemm_kernel<<<(waves + 7) / 8, TPB, 0, stream>>>(
            agg, hB, invdeg, Wl2, Wr2, bl2, (float*)d_out, K, NO, tilesM);
    }
}